// BatchedPrecisionAttentionBlock_v1_7645041787177
// MI455X (gfx1250) — compile-verified
//
#include <hip/hip_runtime.h>

// BatchedPrecisionAttentionBlock for MI455X (gfx1250, wave32).
// B=2, S=512, D=64. fp32 throughout; complex GEMMs on V_WMMA_F32_16X16X4_F32.

#define BB 2
#define SS 512
#define DD 64
#define EPS_C 1e-6f
#define DELTA_C 1.0f
#define ETA_C 1.0f

typedef float v2f __attribute__((ext_vector_type(2)));
typedef float v8f __attribute__((ext_vector_type(8)));

// Only check on the device pass; the host (x86) pass reports __has_builtin==false
// for aux-target builtins even though it parses them fine.
#if defined(__HIP_DEVICE_COMPILE__)
#if !__has_builtin(__builtin_amdgcn_wmma_f32_16x16x4_f32)
#error "wmma_f32_16x16x4_f32 builtin not available on this toolchain (device pass)"
#endif
#endif

// ---------------------------------------------------------------------------
// Stage 0: decay/precision tables indexed by delta = i - j (uniform time grid).
// Etab_r/Etab_i: exp(lam_re*tau)*{cos,sin}(lam_im*tau); Ptab: alpha/(var+eps).
// Also emits lambda_h = [lam_re; lam_im] to d_out.
// ---------------------------------------------------------------------------
__global__ __launch_bounds__(64)
void build_tables(const float* __restrict__ t_all,
                  const float* __restrict__ lambda_r,
                  const float* __restrict__ lambda_i,
                  const float* __restrict__ om_s,
                  const float* __restrict__ om0_s,
                  const float* __restrict__ gam_s,
                  const float* __restrict__ alpha_nu,
                  float* __restrict__ Etab_r,
                  float* __restrict__ Etab_i,
                  float* __restrict__ Ptab,
                  float* __restrict__ lambda_out) {
  const int d = threadIdx.x;       // 0..63
  const int delta = blockIdx.x;    // 0..511
  float lam_re, lam_im;
  if (d < 32) {
    float lr = lambda_r[d];
    lam_re = -lr * lr;
    lam_im = lambda_i[d];
  } else {
    float lr = lambda_r[d - 32];
    lam_re = -lr * lr;
    lam_im = -lambda_i[d - 32];
  }
  const float tau = t_all[delta] - t_all[0];   // t_measure_all[0, delta]-t[0] >= 0
  const float e  = expf(lam_re * tau);
  const float er = e * cosf(lam_im * tau);
  const float ei = e * sinf(lam_im * tau);
  const float exp2 = e * e;
  const float Om  = om_s[d]  * om_s[d];
  const float Om0 = om0_s[d] * om0_s[d];
  const float Ga  = gam_s[d] * gam_s[d];
  const float var = Om * (1.0f - exp2) / (-2.0f * lam_re + EPS_C) + Om0 * exp2 + Ga;
  const float P = alpha_nu[d] / (var + EPS_C);
  const int idx = delta * DD + d;
  Etab_r[idx] = er;
  Etab_i[idx] = ei;
  Ptab[idx]   = P;
  if (delta == 0) {
    lambda_out[d]      = lam_re;
    lambda_out[DD + d] = lam_im;
  }
}

// ---------------------------------------------------------------------------
// Complex GEMM + bias (+ optional fused residual) via V_WMMA_F32_16X16X4_F32.
// Y[b,s,e] = scale * ( sum_d X[b,s,d]*W[e,d] (complex) + Wb[e] )
//            + addScale * Add[b,s,e]                (if Add != null)
// X/Y/Add layout [B,2,S,D] (real plane, imag plane). W: [2,64,64]. Wb: [2,64].
// One wave per 16x16 (s,e) tile; K=64 in 16 steps of 4; 4 WMMAs per step.
// ---------------------------------------------------------------------------
__global__ __launch_bounds__(128)
void cgemm_wmma(const float* __restrict__ X, const float* __restrict__ W,
                const float* __restrict__ Wb, float* __restrict__ Y,
                const float* __restrict__ Add, float scale, float addScale) {
  const int b    = blockIdx.y;
  const int m0   = blockIdx.x * 16;        // S tile
  const int n0   = threadIdx.y * 16;       // D tile (4 waves cover 64)
  const int lane = threadIdx.x;            // 0..31, one wave per y-row

  const float* Xr = X + (size_t)b * 2 * SS * DD;
  const float* Xi = Xr + (size_t)SS * DD;
  const float* Wr = W;                     // W[0][e][d]
  const float* Wi = W + DD * DD;           // W[1][e][d]

  // A fragment (16x4 f32): lanes 0-15 -> K = k0+{0,1}; lanes 16-31 -> K = k0+{2,3}
  // B fragment (4x16 f32): mirrored, N across lanes; B[k][n] = W[n][k] (Y = X * W^T)
  const int mr = lane & 15;
  const int kh = (lane >> 4) * 2;

  const float* arp = Xr + (size_t)(m0 + mr) * DD + kh;
  const float* aip = Xi + (size_t)(m0 + mr) * DD + kh;
  const float* brp = Wr + (size_t)(n0 + mr) * DD + kh;
  const float* bip = Wi + (size_t)(n0 + mr) * DD + kh;

  v8f cr = {0.f, 0.f, 0.f, 0.f, 0.f, 0.f, 0.f, 0.f};
  v8f ci = {0.f, 0.f, 0.f, 0.f, 0.f, 0.f, 0.f, 0.f};

#pragma unroll
  for (int k0 = 0; k0 < DD; k0 += 4) {
    v2f ar = *(const v2f*)(arp + k0);
    v2f ai = *(const v2f*)(aip + k0);
    v2f br = *(const v2f*)(brp + k0);
    v2f bi = *(const v2f*)(bip + k0);
    v2f nbi;
    nbi.x = -bi.x;
    nbi.y = -bi.y;
    // Yr += Ar*Br - Ai*Bi ; Yi += Ar*Bi + Ai*Br
    cr = __builtin_amdgcn_wmma_f32_16x16x4_f32(false, ar, false, br,  (short)0, cr, false, false);
    cr = __builtin_amdgcn_wmma_f32_16x16x4_f32(false, ai, false, nbi, (short)0, cr, false, false);
    ci = __builtin_amdgcn_wmma_f32_16x16x4_f32(false, ar, false, bi,  (short)0, ci, false, false);
    ci = __builtin_amdgcn_wmma_f32_16x16x4_f32(false, ai, false, br,  (short)0, ci, false, false);
  }

  // D layout: lane -> N = n0 + (lane&15); VGPR v -> M = m0 + v + 8*(lane>=16)
  const int col   = n0 + (lane & 15);
  const int rbase = m0 + ((lane >> 4) << 3);
  const float bre = Wb[col];
  const float bim = Wb[DD + col];
  float* Yr = Y + (size_t)b * 2 * SS * DD;
  float* Yi = Yr + (size_t)SS * DD;
  const float* Ar = Add ? (Add + (size_t)b * 2 * SS * DD) : nullptr;
  const float* Ai = Ar ? (Ar + (size_t)SS * DD) : nullptr;
#pragma unroll
  for (int v = 0; v < 8; ++v) {
    const size_t o = (size_t)(rbase + v) * DD + col;
    float yr = scale * (cr[v] + bre);
    float yi = scale * (ci[v] + bim);
    if (Ar) {
      yr += addScale * Ar[o];
      yi += addScale * Ai[o];
    }
    Yr[o] = yr;
    Yi[o] = yi;
  }
}

// ---------------------------------------------------------------------------
// Stage 2: causal attention core. One block per (i, b); 8 waves split j.
// Each lane owns channels d0=lane, d1=lane+32 (wave32). The per-(i,j)
// denominator is a 32-lane shuffle reduction. Writes unnormalized A into the
// Qij slot, X_hat to its slot, zeros for j>i, and accumulates row sums and
// est_v = (sum_j A*Xhat) / (sum_j A).
// ---------------------------------------------------------------------------
__global__ __launch_bounds__(256)
void attn_core(const float* __restrict__ Qm, const float* __restrict__ Km,
               const float* __restrict__ Vm,
               const float* __restrict__ Etab_r, const float* __restrict__ Etab_i,
               const float* __restrict__ Ptab, const float* __restrict__ beta_nu,
               float* __restrict__ Qij, float* __restrict__ Xhat,
               float* __restrict__ rowsum, float* __restrict__ est_v) {
  const int i    = blockIdx.x;
  const int b    = blockIdx.y;
  const int lane = threadIdx.x & 31;
  const int wave = threadIdx.x >> 5;  // 0..7
  const int d0 = lane, d1 = lane + 32;

  __shared__ float sA[8][64], sXr[8][64], sXi[8][64];

  const size_t baseR = (size_t)(b * 2 + 0) * SS * DD;
  const size_t baseI = (size_t)(b * 2 + 1) * SS * DD;

  const float qr0 = Qm[baseR + (size_t)i * DD + d0];
  const float qr1 = Qm[baseR + (size_t)i * DD + d1];
  const float qi0 = Qm[baseI + (size_t)i * DD + d0];
  const float qi1 = Qm[baseI + (size_t)i * DD + d1];
  const float bet0 = beta_nu[d0] * (1.0f / DD);
  const float bet1 = beta_nu[d1] * (1.0f / DD);

  float acA0 = 0.f, acA1 = 0.f, acXr0 = 0.f, acXr1 = 0.f, acXi0 = 0.f, acXi1 = 0.f;

  for (int j = wave; j <= i; j += 8) {
    const int ti = (i - j) * DD;
    const float er0 = Etab_r[ti + d0], er1 = Etab_r[ti + d1];
    const float ei0 = Etab_i[ti + d0], ei1 = Etab_i[ti + d1];
    const float p0  = Ptab[ti + d0],   p1  = Ptab[ti + d1];

    const size_t kr = baseR + (size_t)j * DD;
    const size_t ki = baseI + (size_t)j * DD;
    const float Kr0 = Km[kr + d0], Kr1 = Km[kr + d1];
    const float Ki0 = Km[ki + d0], Ki1 = Km[ki + d1];
    const float Vr0 = Vm[kr + d0], Vr1 = Vm[kr + d1];
    const float Vi0 = Vm[ki + d0], Vi1 = Vm[ki + d1];

    const float Rr0 = qr0 - (er0 * Kr0 - ei0 * Ki0);
    const float Ri0 = qi0 - (er0 * Ki0 + ei0 * Kr0);
    const float Rr1 = qr1 - (er1 * Kr1 - ei1 * Ki1);
    const float Ri1 = qi1 - (er1 * Ki1 + ei1 * Kr1);

    float w = bet0 * (p0 * (Rr0 * Rr0 + Ri0 * Ri0)) +
              bet1 * (p1 * (Rr1 * Rr1 + Ri1 * Ri1));
#pragma unroll
    for (int off = 16; off > 0; off >>= 1) w += __shfl_xor(w, off, 32);
    const float inv = 1.0f / (1.0f + w);

    const float A0 = p0 * inv, A1 = p1 * inv;
    const float Xr0 = er0 * Vr0 - ei0 * Vi0;
    const float Xh0 = er0 * Vi0 + ei0 * Vr0;
    const float Xr1 = er1 * Vr1 - ei1 * Vi1;
    const float Xh1 = er1 * Vi1 + ei1 * Vr1;

    const size_t qidx = (((size_t)b * SS + i) * SS + j) * DD;
    Qij[qidx + d0] = A0;
    Qij[qidx + d1] = A1;
    const size_t xr = ((((size_t)b * 2 + 0) * SS + i) * SS + j) * DD;
    const size_t xi = ((((size_t)b * 2 + 1) * SS + i) * SS + j) * DD;
    Xhat[xr + d0] = Xr0;  Xhat[xr + d1] = Xr1;
    Xhat[xi + d0] = Xh0;  Xhat[xi + d1] = Xh1;

    acA0 += A0;        acA1 += A1;
    acXr0 += A0 * Xr0; acXr1 += A1 * Xr1;
    acXi0 += A0 * Xh0; acXi1 += A1 * Xh1;
  }

  // masked region j > i must be written as zeros (reference multiplies by tril mask)
  for (int j = i + 1 + wave; j < SS; j += 8) {
    const size_t qidx = (((size_t)b * SS + i) * SS + j) * DD;
    Qij[qidx + d0] = 0.f;
    Qij[qidx + d1] = 0.f;
    const size_t xr = ((((size_t)b * 2 + 0) * SS + i) * SS + j) * DD;
    const size_t xi = ((((size_t)b * 2 + 1) * SS + i) * SS + j) * DD;
    Xhat[xr + d0] = 0.f;  Xhat[xr + d1] = 0.f;
    Xhat[xi + d0] = 0.f;  Xhat[xi + d1] = 0.f;
  }

  sA[wave][d0] = acA0;   sA[wave][d1] = acA1;
  sXr[wave][d0] = acXr0; sXr[wave][d1] = acXr1;
  sXi[wave][d0] = acXi0; sXi[wave][d1] = acXi1;
  __syncthreads();

  if (threadIdx.x < 64) {
    const int d = threadIdx.x;
    float rs = 0.f, xr = 0.f, xi = 0.f;
#pragma unroll
    for (int w2 = 0; w2 < 8; ++w2) {
      rs += sA[w2][d];
      xr += sXr[w2][d];
      xi += sXi[w2][d];
    }
    rowsum[((size_t)b * SS + i) * DD + d] = rs;
    const float invr = 1.0f / rs;
    est_v[((size_t)(b * 2 + 0) * SS + i) * DD + d] = xr * invr;
    est_v[((size_t)(b * 2 + 1) * SS + i) * DD + d] = xi * invr;
  }
}

// ---------------------------------------------------------------------------
// Stage 2b: Qij row normalization (Qij /= sum_j A). Masked part is already 0.
// ---------------------------------------------------------------------------
__global__ __launch_bounds__(256)
void qij_norm(float* __restrict__ Qij, const float* __restrict__ rowsum) {
  const int i = blockIdx.x, b = blockIdx.y;
  __shared__ float inv[64];
  if (threadIdx.x < 64)
    inv[threadIdx.x] = 1.0f / rowsum[((size_t)b * SS + i) * DD + threadIdx.x];
  __syncthreads();
  const size_t base = (((size_t)b * SS + i) * SS) * DD;
  const int n = (i + 1) * DD;
  for (int idx = threadIdx.x; idx < n; idx += 256)
    Qij[base + idx] *= inv[idx & 63];
}

// ---------------------------------------------------------------------------
// Stage 3a: one-step phase rotation pv = M (.) est_v, M = table row delta=1
// (tau = t[1]-t[0] = dt exactly).
// ---------------------------------------------------------------------------
__global__ __launch_bounds__(256)
void rot_kernel(const float* __restrict__ est_v, const float* __restrict__ Etab_r,
                const float* __restrict__ Etab_i, float* __restrict__ pv) {
  const int idx = blockIdx.x * blockDim.x + threadIdx.x;  // over B*S*D
  if (idx >= BB * SS * DD) return;
  const int d = idx & 63;
  const int b = idx / (SS * DD);
  const int rem = idx - b * SS * DD;  // s*D + d
  const float Mr = Etab_r[DD + d];
  const float Mi = Etab_i[DD + d];
  const size_t ro = (size_t)(b * 2 + 0) * SS * DD + rem;
  const size_t io = (size_t)(b * 2 + 1) * SS * DD + rem;
  const float er = est_v[ro];
  const float ei = est_v[io];
  pv[ro] = Mr * er - Mi * ei;
  pv[io] = Mr * ei + Mi * er;
}

// ---------------------------------------------------------------------------
extern "C" void kernel_launch(void* const* d_in, const int* in_sizes, int n_in,
                              void* d_out, int out_size, void* d_ws, size_t ws_size,
                              hipStream_t stream) {
  (void)in_sizes; (void)n_in; (void)out_size; (void)ws_size;
  const float* X_q  = (const float*)d_in[0];
  const float* X_k  = (const float*)d_in[1];
  const float* X_v  = (const float*)d_in[2];
  const float* t_all = (const float*)d_in[3];
  const float* lam_r = (const float*)d_in[4];
  const float* lam_i = (const float*)d_in[5];
  const float* om_s  = (const float*)d_in[6];
  const float* om0_s = (const float*)d_in[7];
  const float* gam_s = (const float*)d_in[8];
  const float* alpha = (const float*)d_in[9];
  const float* beta  = (const float*)d_in[10];
  const float* W_q = (const float*)d_in[11];
  const float* W_k = (const float*)d_in[12];
  const float* W_v = (const float*)d_in[13];
  const float* W_p = (const float*)d_in[14];
  const float* W_e = (const float*)d_in[15];
  const float* W_q_b = (const float*)d_in[16];
  const float* W_k_b = (const float*)d_in[17];
  const float* W_v_b = (const float*)d_in[18];
  const float* W_p_b = (const float*)d_in[19];
  const float* W_e_b = (const float*)d_in[20];

  float* out = (float*)d_out;
  const size_t nBSD2 = (size_t)BB * 2 * SS * DD;       // 131072
  float* out_est    = out;                              // [B,2,S,D]
  float* out_out    = out + nBSD2;                      // [B,2,S,D]
  float* out_Qij    = out + 2 * nBSD2;                  // [B,S,S,D]
  float* out_Xhat   = out_Qij + (size_t)BB * SS * SS * DD;   // [B,2,S,S,D]
  float* out_lambda = out_Xhat + (size_t)BB * 2 * SS * SS * DD;  // [2,D]

  float* ws = (float*)d_ws;
  float* Etab_r = ws;                 // 512*64
  float* Etab_i = Etab_r + SS * DD;
  float* Ptab   = Etab_i + SS * DD;
  float* Qbuf   = Ptab + SS * DD;     // [B,2,S,D]
  float* Kbuf   = Qbuf + nBSD2;
  float* Vbuf   = Kbuf + nBSD2;
  float* rowsum = Vbuf + nBSD2;       // [B,S,D]
  float* est_v  = rowsum + (size_t)BB * SS * DD;  // [B,2,S,D]
  float* pv     = est_v + nBSD2;      // [B,2,S,D]

  // Stage 0: tables + lambda_h output
  build_tables<<<dim3(SS), dim3(DD), 0, stream>>>(
      t_all, lam_r, lam_i, om_s, om0_s, gam_s, alpha,
      Etab_r, Etab_i, Ptab, out_lambda);

  // Stage 1: complex projections on the fp32 WMMA pipe
  const dim3 gG(SS / 16, BB), bG(32, 4);
  cgemm_wmma<<<gG, bG, 0, stream>>>(X_q, W_q, W_q_b, Qbuf, nullptr, 1.0f, 0.0f);
  cgemm_wmma<<<gG, bG, 0, stream>>>(X_k, W_k, W_k_b, Kbuf, nullptr, 1.0f, 0.0f);
  cgemm_wmma<<<gG, bG, 0, stream>>>(X_v, W_v, W_v_b, Vbuf, nullptr, 1.0f, 0.0f);

  // Stage 2: causal core (writes unnormalized Qij, Xhat, rowsums, est_v)
  attn_core<<<dim3(SS, BB), dim3(256), 0, stream>>>(
      Qbuf, Kbuf, Vbuf, Etab_r, Etab_i, Ptab, beta,
      out_Qij, out_Xhat, rowsum, est_v);

  // Stage 2b: normalize Qij rows
  qij_norm<<<dim3(SS, BB), dim3(256), 0, stream>>>(out_Qij, rowsum);

  // Stage 3: est = W_e(est_v)+b ; out = DELTA*(W_p(M.est_v)+b) + ETA*est
  cgemm_wmma<<<gG, bG, 0, stream>>>(est_v, W_e, W_e_b, out_est, nullptr, 1.0f, 0.0f);
  rot_kernel<<<dim3((BB * SS * DD + 255) / 256), dim3(256), 0, stream>>>(
      est_v, Etab_r, Etab_i, pv);
  cgemm_wmma<<<gG, bG, 0, stream>>>(pv, W_p, W_p_b, out_out, out_est, DELTA_C, ETA_C);
}